// GCNLayer_60009283059862
// MI455X (gfx1250) — compile-verified
//
#include <hip/hip_runtime.h>
#include <hip/hip_bf16.h>

typedef __attribute__((ext_vector_type(2))) float v2f;
typedef __attribute__((ext_vector_type(8))) float v8f;

#define ROWPAD 72  // 64 data floats + 8 pad: breaks stride-64 bank conflicts, keeps rows 16B-aligned

// ---------------------------------------------------------------------------
// Kernel 1: zero the accumulator workspace (N*64 floats) with 128-bit stores.
// ---------------------------------------------------------------------------
__global__ void gcn_zero_kernel(float4* __restrict__ p, long n4) {
    long i = (long)blockIdx.x * blockDim.x + threadIdx.x;
    if (i < n4) {
        p[i] = make_float4(0.f, 0.f, 0.f, 0.f);
    }
}

// ---------------------------------------------------------------------------
// Kernel 2: edge scatter.  One wave32 per edge; lane l owns feature elements
// [2l, 2l+1].  feature[src] gather is a contiguous 256B row per wave
// (coalesced), atomics land contiguously on accum[dst] (resolved in L2).
// ---------------------------------------------------------------------------
__global__ void gcn_scatter_kernel(const float* __restrict__ feature,
                                   const float* __restrict__ norm,
                                   const int* __restrict__ src,
                                   const int* __restrict__ dst,
                                   float* __restrict__ accum,
                                   int E) {
    long t   = (long)blockIdx.x * blockDim.x + threadIdx.x;
    int  e   = (int)(t >> 5);   // 32 lanes per edge (wave32)
    int  lane = (int)(t & 31);
    if (e >= E) return;

    int   s   = src[e];
    int   d   = dst[e];
    float nrm = norm[s];

    const float2* f2 = reinterpret_cast<const float2*>(feature);
    float2 v = f2[(long)s * 32 + lane];

    float* out = accum + (long)d * 64 + lane * 2;
    atomicAdd(out + 0, v.x * nrm);
    atomicAdd(out + 1, v.y * nrm);
}

// ---------------------------------------------------------------------------
// Kernel 3: out = (accum * norm) @ W^T + b  via V_WMMA_F32_16X16X4_F32.
// 256 threads = 8 waves; each wave owns a 16-row tile (block covers 128 rows).
// W (64x64) staged once per block into LDS; each wave stages its 16 rows of
// accum*norm into LDS.  Per wave: 4 n-tiles x 16 k-steps = 64 fp32 WMMAs.
// ---------------------------------------------------------------------------
__global__ void gcn_wmma_gemm_kernel(const float* __restrict__ accum,
                                     const float* __restrict__ norm,
                                     const float* __restrict__ W,
                                     const float* __restrict__ bias,
                                     float* __restrict__ out,
                                     int N) {
    __shared__ float ldsW[64 * ROWPAD];         // W row-major, padded rows
    __shared__ float ldsA[8 * 16 * ROWPAD];     // per-wave 16x64 A tiles

    const int tid  = threadIdx.x;   // 0..255
    const int wave = tid >> 5;      // 0..7
    const int lane = tid & 31;

    // ---- stage W[64][64] -> LDS (coalesced float4 loads, 16B-aligned stores)
    {
        const float4* W4 = reinterpret_cast<const float4*>(W);
        #pragma unroll
        for (int i = 0; i < 4; ++i) {
            int idx = i * 256 + tid;        // float4 index 0..1023
            int row = idx >> 4;             // 16 float4 per 64-float row
            int c4  = idx & 15;
            float4 v = W4[idx];
            *reinterpret_cast<float4*>(&ldsW[row * ROWPAD + c4 * 4]) = v;
        }
    }

    const long r0 = (long)blockIdx.x * 128 + (long)wave * 16;

    // ---- stage this wave's 16 rows of (accum * norm) -> LDS
    {
        const float4* A4 = reinterpret_cast<const float4*>(accum);
        float* myA = &ldsA[wave * 16 * ROWPAD];
        #pragma unroll
        for (int i = 0; i < 8; ++i) {
            int idx = i * 32 + lane;        // 0..255 : row=idx/16, col4=idx%16
            int row = idx >> 4;
            int c4  = idx & 15;
            long gr  = r0 + row;
            long grc = (gr < (long)N) ? gr : (long)(N - 1);  // clamp, no EXEC divergence
            float4 v = A4[grc * 16 + c4];
            float  nm = norm[grc];
            v.x *= nm; v.y *= nm; v.z *= nm; v.w *= nm;
            *reinterpret_cast<float4*>(&myA[row * ROWPAD + c4 * 4]) = v;
        }
    }
    __syncthreads();

    const float* myA     = &ldsA[wave * 16 * ROWPAD];
    const int    halfsel = lane >> 4;   // 0: K even pair base, 1: +2
    const int    l15     = lane & 15;

    v8f acc[4] = {v8f{0}, v8f{0}, v8f{0}, v8f{0}};

    // D = A(16x64) x Wt(64x64): loop K in chunks of 4, 4 output n-tiles of 16
    #pragma unroll
    for (int kk = 0; kk < 16; ++kk) {
        const int kb = kk * 4 + halfsel * 2;
        // A fragment (16x4 fp32): lanes 0-15 hold K=kb..kb+1 for M=lane,
        // lanes 16-31 hold K=kb+2..kb+3 (halfsel folds this into kb).
        v2f a;
        a.x = myA[l15 * ROWPAD + kb + 0];
        a.y = myA[l15 * ROWPAD + kb + 1];
        #pragma unroll
        for (int nt = 0; nt < 4; ++nt) {
            const int n = nt * 16 + l15;
            // B fragment (4x16 fp32): Wt[k][n] = W[n][k]
            v2f bf;
            bf.x = ldsW[n * ROWPAD + kb + 0];
            bf.y = ldsW[n * ROWPAD + kb + 1];
            acc[nt] = __builtin_amdgcn_wmma_f32_16x16x4_f32(
                /*neg_a=*/false, a, /*neg_b=*/false, bf,
                /*c_mod=*/(short)0, acc[nt],
                /*reuse_a=*/false, /*reuse_b=*/false);
        }
    }

    // ---- store D tiles + bias.  C/D layout: VGPR v -> M = v + 8*halfsel,
    // N = l15 (per 16-wide n-tile).  Mask only the stores for tail rows.
    #pragma unroll
    for (int nt = 0; nt < 4; ++nt) {
        const int  nc = nt * 16 + l15;
        const float bv = bias[nc];
        #pragma unroll
        for (int v = 0; v < 8; ++v) {
            long m = r0 + v + halfsel * 8;
            if (m < (long)N) {
                out[m * 64 + nc] = acc[nt][v] + bv;
            }
        }
    }
}

// ---------------------------------------------------------------------------
// Launch: inputs in setup_inputs() order:
//   0: feature [N,64] f32   1: norm [N,1] f32   2: src [E] i32
//   3: dst [E] i32          4: W [64,64] f32    5: b [64] f32
// d_out: [N,64] f32.  d_ws: accum [N,64] f32 (zeroed every call).
// ---------------------------------------------------------------------------
extern "C" void kernel_launch(void* const* d_in, const int* in_sizes, int n_in,
                              void* d_out, int out_size, void* d_ws, size_t ws_size,
                              hipStream_t stream) {
    const float* feature = (const float*)d_in[0];
    const float* norm    = (const float*)d_in[1];
    const int*   src     = (const int*)d_in[2];
    const int*   dst     = (const int*)d_in[3];
    const float* W       = (const float*)d_in[4];
    const float* bias    = (const float*)d_in[5];
    float*       out     = (float*)d_out;
    float*       accum   = (float*)d_ws;

    const int N = in_sizes[1];      // norm has one element per node
    const int E = in_sizes[2];

    // 1) zero accumulator
    {
        long n4 = (long)N * 16;     // N*64 floats / 4
        long blocks = (n4 + 255) / 256;
        gcn_zero_kernel<<<(unsigned)blocks, 256, 0, stream>>>(
            reinterpret_cast<float4*>(accum), n4);
    }

    // 2) edge scatter (one wave32 per edge)
    {
        long threads = (long)E * 32;
        long blocks  = (threads + 255) / 256;
        gcn_scatter_kernel<<<(unsigned)blocks, 256, 0, stream>>>(
            feature, norm, src, dst, accum, E);
    }

    // 3) WMMA GEMM + bias (128 rows per block)
    {
        long blocks = ((long)N + 127) / 128;
        gcn_wmma_gemm_kernel<<<(unsigned)blocks, 256, 0, stream>>>(
            accum, norm, W, bias, out, N);
    }
}